// Embedding_85401129713725
// MI455X (gfx1250) — compile-verified
//
#include <hip/hip_runtime.h>

// Problem constants (match reference)
constexpr int N_SAMPLES = 65536;
constexpr int DIM       = 1024;
constexpr int CHANNELS  = 128;

constexpr int SPLITS         = 64;                   // row splits for phase-1 partials
constexpr int ROWS_PER_SPLIT = N_SAMPLES / SPLITS;   // 1024

typedef __attribute__((ext_vector_type(2))) float v2f;
typedef __attribute__((ext_vector_type(8))) float v8f;

// Workspace layout (in floats)
constexpr int WS_PART = 0;                    // SPLITS * DIM partial column sums of x
constexpr int WS_C    = SPLITS * DIM;         // DIM     : c = sum(x, axis=0)
constexpr int WS_T    = WS_C + DIM;           // DIM     : t = sum(w, axis=0)
constexpr int WS_S    = WS_T + DIM;           // CHANNELS: s = sum(w, axis=1)

// ---------------------------------------------------------------------------
// Phase 1: streaming column-sum of x via V_WMMA_F32_16X16X4_F32.
//   A = ones(16x4)  ->  D[m,n] = sum_k B[k,n] + C[m,n]  (all rows identical).
// Each wave owns 16 columns and a 1024-row slice; B tiles are 4 rows x 16 cols.
// Non-temporal loads: x (256 MB) is a one-shot stream, keep it out of L2.
// ---------------------------------------------------------------------------
__global__ void colsum_x_wmma(const float* __restrict__ x,
                              float* __restrict__ part) {
    const int lane  = threadIdx.x & 31;
    const int wave  = threadIdx.x >> 5;
    const int col0  = blockIdx.x * 64 + wave * 16;     // 4 waves * 16 cols = 64 cols/block
    const int split = blockIdx.y;
    const int r0    = split * ROWS_PER_SPLIT;

    const int col   = col0 + (lane & 15);
    const int khalf = (lane >> 4) << 1;                // lanes 0-15: K=0/1, lanes 16-31: K=2/3

    v2f a; a.x = 1.0f; a.y = 1.0f;                     // A = ones(16x4), layout-invariant
    v8f acc = {};

    const float* base = x + (size_t)r0 * DIM + col;
    #pragma unroll 4
    for (int r = 0; r < ROWS_PER_SPLIT; r += 4) {
        const float* p = base + (size_t)r * DIM;
        v2f b;
        b.x = __builtin_nontemporal_load(p + (size_t)khalf * DIM);        // rows r+{0|2}
        b.y = __builtin_nontemporal_load(p + (size_t)(khalf + 1) * DIM);  // rows r+{1|3}
        // (neg_a, A, neg_b, B, c_mod, C, reuse_a, reuse_b)
        acc = __builtin_amdgcn_wmma_f32_16x16x4_f32(
            false, a, false, b, (short)0, acc, false, false);
    }
    // D row M=0 lives in acc[0], lanes 0..15 hold N=0..15 (per 32-bit C/D layout)
    if (lane < 16) part[(size_t)split * DIM + col] = acc[0];
}

// ---------------------------------------------------------------------------
// Phase 2: deterministic reduction of the 64 partials per column -> c[DIM]
// ---------------------------------------------------------------------------
__global__ void reduce_parts(const float* __restrict__ part,
                             float* __restrict__ c) {
    const int col = blockIdx.x * blockDim.x + threadIdx.x;
    float s = 0.0f;
    #pragma unroll 8
    for (int k = 0; k < SPLITS; ++k) s += part[(size_t)k * DIM + col];
    c[col] = s;
}

// ---------------------------------------------------------------------------
// Phase 3: weight reductions. Blocks 0..3: t[j] = sum_i w[i,j] (coalesced).
//          Block 4: s[i] = sum_j w[i,:]. Weight is only 512 KB; stays in L2.
// ---------------------------------------------------------------------------
__global__ void weight_sums(const float* __restrict__ w,
                            float* __restrict__ t,
                            float* __restrict__ s) {
    if (blockIdx.x < 4) {
        const int j = blockIdx.x * 256 + threadIdx.x;
        float acc = 0.0f;
        #pragma unroll 4
        for (int i = 0; i < CHANNELS; ++i) acc += w[(size_t)i * DIM + j];
        t[j] = acc;
    } else {
        const int i = threadIdx.x;
        if (i < CHANNELS) {
            const float* row = w + (size_t)i * DIM;
            float acc = 0.0f;
            #pragma unroll 4
            for (int j = 0; j < DIM; ++j) acc += row[j];
            s[i] = acc;
        }
    }
}

// ---------------------------------------------------------------------------
// Phase 4: one block per channel i.
//   det_i = sum_j exp(s_i * c_j)   (LDS tree reduce, v_exp_f32)
//   out[i,j] = exp(s_i * t_j) / det_i
// ---------------------------------------------------------------------------
__global__ void softreorder_out(const float* __restrict__ cvec,
                                const float* __restrict__ tvec,
                                const float* __restrict__ svec,
                                float* __restrict__ out) {
    __shared__ float red[256];
    const int i  = blockIdx.x;
    const float si = svec[i];

    float p = 0.0f;
    for (int j = threadIdx.x; j < DIM; j += 256) p += __expf(si * cvec[j]);
    red[threadIdx.x] = p;
    __syncthreads();
    #pragma unroll
    for (int off = 128; off > 0; off >>= 1) {
        if (threadIdx.x < off) red[threadIdx.x] += red[threadIdx.x + off];
        __syncthreads();
    }
    const float inv_det = 1.0f / red[0];

    float* orow = out + (size_t)i * DIM;
    for (int j = threadIdx.x; j < DIM; j += 256)
        orow[j] = __expf(si * tvec[j]) * inv_det;
}

// ---------------------------------------------------------------------------
extern "C" void kernel_launch(void* const* d_in, const int* in_sizes, int n_in,
                              void* d_out, int out_size, void* d_ws, size_t ws_size,
                              hipStream_t stream) {
    (void)in_sizes; (void)n_in; (void)out_size; (void)ws_size;
    const float* x = (const float*)d_in[0];   // [N_SAMPLES, DIM]
    const float* w = (const float*)d_in[1];   // [CHANNELS, DIM]
    float* ws   = (float*)d_ws;
    float* part = ws + WS_PART;
    float* c    = ws + WS_C;
    float* t    = ws + WS_T;
    float* s    = ws + WS_S;
    float* out  = (float*)d_out;              // [CHANNELS, DIM]

    // Phase 1: 16 column-groups x 64 row-splits, 128 threads (4 wave32) each.
    colsum_x_wmma<<<dim3(DIM / 64, SPLITS), 128, 0, stream>>>(x, part);
    // Phase 2: finish c.
    reduce_parts<<<dim3(DIM / 256), 256, 0, stream>>>(part, c);
    // Phase 3: s and t from weight.
    weight_sums<<<5, 256, 0, stream>>>(w, t, s);
    // Phase 4: det + output.
    softreorder_out<<<CHANNELS, 256, 0, stream>>>(c, t, s, out);
}